// EdgeBiasMLP_31928786879006
// MI455X (gfx1250) — compile-verified
//
#include <hip/hip_runtime.h>
#include <cmath>

typedef float v2f __attribute__((ext_vector_type(2)));
typedef float v4f __attribute__((ext_vector_type(4)));
typedef float v8f __attribute__((ext_vector_type(8)));

// ---------------------------------------------------------------------------
// K0: zero-fill output with wide non-temporal stores (bandwidth-bound phase).
// ---------------------------------------------------------------------------
__global__ void ebm_zero_f32(float* __restrict__ p, size_t n) {
    size_t tid    = (size_t)blockIdx.x * blockDim.x + threadIdx.x;
    size_t stride = (size_t)gridDim.x * blockDim.x;
    size_t n4     = n >> 2;
    v4f z = {0.0f, 0.0f, 0.0f, 0.0f};
    for (size_t i = tid; i < n4; i += stride)
        __builtin_nontemporal_store(z, ((v4f*)p) + i);
    // scalar tail
    for (size_t i = (n4 << 2) + tid; i < n; i += stride)
        p[i] = 0.0f;
}

// ---------------------------------------------------------------------------
// K1: clear owner[] only at slots touched by edges (4 MiB instead of 64 MiB).
// ---------------------------------------------------------------------------
__global__ void ebm_owner_clear(const int* __restrict__ ei,
                                unsigned* __restrict__ owner, int E, int N) {
    int e = blockIdx.x * blockDim.x + threadIdx.x;
    if (e < E) {
        int u = ei[e], v = ei[E + e];
        owner[(size_t)u * N + v] = 0u;
    }
}

// ---------------------------------------------------------------------------
// K2: deterministic last-write-wins: highest edge index owns the slot.
// ---------------------------------------------------------------------------
__global__ void ebm_owner_mark(const int* __restrict__ ei,
                               unsigned* __restrict__ owner, int E, int N) {
    int e = blockIdx.x * blockDim.x + threadIdx.x;
    if (e < E) {
        int u = ei[e], v = ei[E + e];
        atomicMax(&owner[(size_t)u * N + v], (unsigned)(e + 1));
    }
}

// ---------------------------------------------------------------------------
// K3: WMMA f32 GEMM (16-edge tile per wave, K=128 via 32x v_wmma_f32_16x16x4)
//     + scatter of the 16x8 result into out[h, u, v].
//     W is staged transposed & padded to 16 cols in LDS (stride 130: avoids
//     16-way bank conflicts, keeps 8B alignment for b64 LDS loads).
// ---------------------------------------------------------------------------
#define WT_STRIDE 130

template <bool DEDUP>
__global__ void ebm_gemm_scatter(const int*      __restrict__ ei,
                                 const float*    __restrict__ A,     // [E,128]
                                 const float*    __restrict__ W,     // [128,8]
                                 const float*    __restrict__ bias,  // [8]
                                 const unsigned* __restrict__ owner,
                                 float*          __restrict__ out,   // [8,N,N]
                                 int E, int N) {
    __shared__ float Wt[16 * WT_STRIDE];  // Wt[n][k] = W[k][n], n>=8 -> 0

    for (int idx = threadIdx.x; idx < 16 * 128; idx += blockDim.x) {
        int n = idx >> 7, k = idx & 127;
        Wt[n * WT_STRIDE + k] = (n < 8) ? W[k * 8 + n] : 0.0f;
    }
    __syncthreads();

    const int lane = threadIdx.x & 31;
    const int wave = threadIdx.x >> 5;
    const int tile = blockIdx.x * (blockDim.x >> 5) + wave;
    const int base = tile * 16;
    if (base >= E) return;  // wave-uniform: EXEC stays all-ones for WMMA

    const int r  = lane & 15;   // A: row-in-tile   B: head column
    const int hi = lane >> 4;   // selects K pair {0,1} vs {2,3}

    int m = base + r;
    if (m >= E) m = E - 1;      // clamp A reads for ragged last tile
    const float* arow = A  + (size_t)m * 128 + 2 * hi;
    const float* wrow = Wt + r * WT_STRIDE + 2 * hi;

    v8f c = {};
#pragma unroll
    for (int k0 = 0; k0 < 128; k0 += 4) {
        v2f a  = *(const v2f*)(arow + k0);   // A[m][k0+2*hi .. +1]
        v2f bb = *(const v2f*)(wrow + k0);   // B[k0+2*hi .. +1][r]
        c = __builtin_amdgcn_wmma_f32_16x16x4_f32(
                false, a, false, bb, (short)0, c, false, false);
    }

    // D layout: VGPR i, lanes 0-15 -> score[edge base+i   ][head lane],
    //           lanes 16-31        -> score[edge base+8+i ][head lane-16]
    const int head = r;
    if (head < 8) {
        const float  bh    = bias[head];
        const int    ebase = base + hi * 8;
        const size_t NN    = (size_t)N * N;
        float* outh = out + (size_t)head * NN;
#pragma unroll
        for (int i = 0; i < 8; ++i) {
            int e = ebase + i;
            if (e < E) {
                int u = ei[e], v = ei[E + e];
                size_t slot = (size_t)u * N + v;
                float  val  = c[i] + bh;
                if (!DEDUP || owner[slot] == (unsigned)(e + 1))
                    outh[slot] = val;
            }
        }
    }
}

// ---------------------------------------------------------------------------
// Generic scalar fallback (any D/H) — keeps launcher correct off-shape.
// ---------------------------------------------------------------------------
__global__ void ebm_fallback(const int* __restrict__ ei,
                             const float* __restrict__ A,
                             const float* __restrict__ W,
                             const float* __restrict__ bias,
                             const unsigned* __restrict__ owner, int dedup,
                             float* __restrict__ out,
                             int E, int D, int H, int N) {
    long long idx = (long long)blockIdx.x * blockDim.x + threadIdx.x;
    if (idx >= (long long)E * H) return;
    int e = (int)(idx / H), h = (int)(idx % H);
    float s = bias[h];
    for (int k = 0; k < D; ++k) s += A[(size_t)e * D + k] * W[(size_t)k * H + h];
    int u = ei[e], v = ei[E + e];
    size_t slot = (size_t)u * N + v;
    if (!dedup || owner[slot] == (unsigned)(e + 1))
        out[(size_t)h * N * N + slot] = s;
}

extern "C" void kernel_launch(void* const* d_in, const int* in_sizes, int n_in,
                              void* d_out, int out_size, void* d_ws, size_t ws_size,
                              hipStream_t stream) {
    const int*   ei   = (const int*)d_in[0];
    const float* attr = (const float*)d_in[1];
    const float* W    = (const float*)d_in[2];
    const float* bias = (const float*)d_in[3];
    float*       out  = (float*)d_out;

    const int E = in_sizes[0] / 2;
    const int D = (E > 0) ? in_sizes[1] / E : 0;
    const int H = in_sizes[3];
    const long long nn = (H > 0) ? (long long)out_size / H : 0;
    const int N = (int)llround(sqrt((double)nn));
    if (E <= 0 || H <= 0 || N <= 0) return;

    // K0: zero the 512 MiB output.
    ebm_zero_f32<<<8192, 256, 0, stream>>>(out, (size_t)out_size);

    // K1/K2: deterministic duplicate resolution if scratch is big enough.
    const size_t need  = (size_t)N * (size_t)N * sizeof(unsigned);
    const bool   dedup = (d_ws != nullptr) && (ws_size >= need);
    unsigned* owner = (unsigned*)d_ws;
    if (dedup) {
        int eb = (E + 255) / 256;
        ebm_owner_clear<<<eb, 256, 0, stream>>>(ei, owner, E, N);
        ebm_owner_mark <<<eb, 256, 0, stream>>>(ei, owner, E, N);
    }

    // K3: WMMA GEMM + scatter (fast path), scalar fallback otherwise.
    if (D == 128 && H == 8) {
        int tiles  = (E + 15) / 16;
        int blocks = (tiles + 7) / 8;  // 8 waves (256 threads) per block
        if (dedup)
            ebm_gemm_scatter<true ><<<blocks, 256, 0, stream>>>(
                ei, attr, W, bias, owner, out, E, N);
        else
            ebm_gemm_scatter<false><<<blocks, 256, 0, stream>>>(
                ei, attr, W, bias, owner, out, E, N);
    } else {
        long long total = (long long)E * H;
        int blocks = (int)((total + 255) / 256);
        ebm_fallback<<<blocks, 256, 0, stream>>>(
            ei, attr, W, bias, owner, dedup ? 1 : 0, out, E, D, H, N);
    }
}